// Model_17944373363339
// MI455X (gfx1250) — compile-verified
//
#include <hip/hip_runtime.h>
#include <hip/hip_bf16.h>

#define USER 100000
#define ITEM 50000
#define NTOT (USER + ITEM)
#define D 64

typedef __attribute__((ext_vector_type(2))) float v2f;
typedef __attribute__((ext_vector_type(8))) float v8f;

// ---------------- zero ----------------
__global__ void zero_kernel(float* __restrict__ p, size_t n) {
    size_t n4 = n >> 2;
    size_t i = (size_t)blockIdx.x * blockDim.x + threadIdx.x;
    size_t stride = (size_t)gridDim.x * blockDim.x;
    float4* p4 = (float4*)p;
    for (; i < n4; i += stride) p4[i] = make_float4(0.f, 0.f, 0.f, 0.f);
}

// ---------------- WMMA fp32 GEMM: out(M x 64) = A(M x K) @ W(K x 64) + bias ----------------
// One wave per 16-row tile. 4 accumulators cover N=64. K stepped by 4 via V_WMMA_F32_16X16X4_F32.
__global__ __launch_bounds__(256) void gemm_wmma_f32(
    const float* __restrict__ A, const float* __restrict__ W,
    const float* __restrict__ bias, float* __restrict__ out,
    int M, int K) {
    int wave = (int)(((size_t)blockIdx.x * blockDim.x + threadIdx.x) >> 5);
    int lane = threadIdx.x & 31;
    int ntiles = M >> 4;
    if (wave >= ntiles) return;          // wave-uniform: EXEC stays all-ones for WMMA

    int m0 = wave << 4;
    int half = lane >> 4;                // 0 or 1 (lane half)
    int n = lane & 15;
    int khalf = half << 1;               // A/B K-offset for this lane half: 0 or 2

    const float* arow = A + (size_t)(m0 + n) * K + khalf;   // A: lanes hold M=lane&15
    const float* wb = W + (size_t)khalf * D + n;            // B: lanes hold N=lane&15

    v8f acc0 = {}, acc1 = {}, acc2 = {}, acc3 = {};
    for (int k0 = 0; k0 < K; k0 += 4) {
        v2f a;
        a.x = arow[k0];
        a.y = arow[k0 + 1];
        const float* wk = wb + (size_t)k0 * D;
        v2f b0; b0.x = wk[0];  b0.y = wk[D + 0];
        v2f b1; b1.x = wk[16]; b1.y = wk[D + 16];
        v2f b2; b2.x = wk[32]; b2.y = wk[D + 32];
        v2f b3; b3.x = wk[48]; b3.y = wk[D + 48];
        acc0 = __builtin_amdgcn_wmma_f32_16x16x4_f32(false, a, false, b0, (short)0, acc0, false, false);
        acc1 = __builtin_amdgcn_wmma_f32_16x16x4_f32(false, a, false, b1, (short)0, acc1, false, false);
        acc2 = __builtin_amdgcn_wmma_f32_16x16x4_f32(false, a, false, b2, (short)0, acc2, false, false);
        acc3 = __builtin_amdgcn_wmma_f32_16x16x4_f32(false, a, false, b3, (short)0, acc3, false, false);
    }

    float bb0 = bias[n], bb1 = bias[16 + n], bb2 = bias[32 + n], bb3 = bias[48 + n];
#pragma unroll
    for (int r = 0; r < 8; ++r) {
        int row = m0 + r + (half << 3);  // C/D layout: VGPR r -> M=r (lanes 0-15) / M=8+r (16-31)
        float* o = out + (size_t)row * D + n;
        o[0]  = acc0[r] + bb0;
        o[16] = acc1[r] + bb1;
        o[32] = acc2[r] + bb2;
        o[48] = acc3[r] + bb3;
    }
}

// ---------------- row-wise L2 normalize in place ----------------
__global__ void l2norm_rows(float* __restrict__ x, int nrows) {
    int i = blockIdx.x * blockDim.x + threadIdx.x;
    if (i >= nrows) return;
    float* p = x + (size_t)i * D;
    float s = 0.f;
#pragma unroll
    for (int j = 0; j < D; ++j) s += p[j] * p[j];
    float sc = 1.0f / fmaxf(sqrtf(s), 1e-12f);
#pragma unroll
    for (int j = 0; j < D; ++j) p[j] *= sc;
}

// ---------------- SpMM: y[rows[e]] += vals[e] * x[cols[e]], x = concat(lo, hi) split at USER ----
__global__ __launch_bounds__(256) void spmm_atomic(
    const int* __restrict__ rows, const int* __restrict__ cols,
    const float* __restrict__ vals,
    const float* __restrict__ xlo, const float* __restrict__ xhi,
    float* __restrict__ y, long long nnz) {
    long long t = (long long)blockIdx.x * blockDim.x + threadIdx.x;
    long long e = t >> 4;
    if (e >= nnz) return;
    int sub = (int)(t & 15);
    if (sub == 0 && e + 4096 < nnz) {   // stream-ahead on the COO arrays
        __builtin_prefetch(cols + e + 4096, 0, 1);
        __builtin_prefetch(vals + e + 4096, 0, 1);
    }
    int r = rows[e];
    int c = cols[e];
    float v = vals[e];
    const float* xr = (c < USER) ? (xlo + (size_t)c * D)
                                 : (xhi + (size_t)(c - USER) * D);
    const float4 xv = *(const float4*)(xr + sub * 4);
    float* yp = y + (size_t)r * D + sub * 4;
    atomicAdd(yp + 0, v * xv.x);
    atomicAdd(yp + 1, v * xv.y);
    atomicAdd(yp + 2, v * xv.z);
    atomicAdd(yp + 3, v * xv.w);
}

// ---------------- elementwise: a = a + b + 0.2*c ----------------
__global__ void fuse3_kernel(float* __restrict__ a, const float* __restrict__ b,
                             const float* __restrict__ c, size_t n) {
    size_t i = (size_t)blockIdx.x * blockDim.x + threadIdx.x;
    if (i >= n) return;
    a[i] = a[i] + b[i] + 0.2f * c[i];
}

// ---------------- attention-weighted modal fusion ----------------
__global__ void modal_fuse(const float* __restrict__ ai, const float* __restrict__ at,
                           const float* __restrict__ A, const float* __restrict__ B,
                           float* __restrict__ M, size_t n) {
    size_t idx = (size_t)blockIdx.x * blockDim.x + threadIdx.x;
    if (idx >= n) return;
    size_t i = idx >> 6;
    float wi = ai[i], wt = at[i];
    float w = wi + wt;
    w = (w == 0.f) ? 1.f : w;
    M[idx] = (wi / w) * A[idx] + (wt / w) * B[idx];
}

// ---------------- elementwise: d += s ----------------
__global__ void add_inplace(float* __restrict__ d, const float* __restrict__ s, size_t n) {
    size_t i = (size_t)blockIdx.x * blockDim.x + threadIdx.x;
    if (i >= n) return;
    d[i] += s[i];
}

// ---------------- out += 0.5 * l2norm(M) row-wise ----------------
__global__ void final_addnorm(const float* __restrict__ Mm, float* __restrict__ out, int nrows) {
    int i = blockIdx.x * blockDim.x + threadIdx.x;
    if (i >= nrows) return;
    const float* p = Mm + (size_t)i * D;
    float* o = out + (size_t)i * D;
    float s = 0.f;
#pragma unroll
    for (int j = 0; j < D; ++j) s += p[j] * p[j];
    float sc = 0.5f / fmaxf(sqrtf(s), 1e-12f);
#pragma unroll
    for (int j = 0; j < D; ++j) o[j] += sc * p[j];
}

extern "C" void kernel_launch(void* const* d_in, const int* in_sizes, int n_in,
                              void* d_out, int out_size, void* d_ws, size_t ws_size,
                              hipStream_t stream) {
    const int*   adj_rows = (const int*)d_in[0];
    const int*   adj_cols = (const int*)d_in[1];
    const float* adj_vals = (const float*)d_in[2];
    const int*   img_rows = (const int*)d_in[3];
    const int*   img_cols = (const int*)d_in[4];
    const float* img_vals = (const float*)d_in[5];
    const int*   txt_rows = (const int*)d_in[6];
    const int*   txt_cols = (const int*)d_in[7];
    const float* txt_vals = (const float*)d_in[8];
    const float* att_image = (const float*)d_in[9];
    const float* att_text  = (const float*)d_in[10];
    const float* uE   = (const float*)d_in[11];
    const float* iE   = (const float*)d_in[12];
    const float* imgE = (const float*)d_in[13];
    const float* txtE = (const float*)d_in[14];
    const float* W_img = (const float*)d_in[15];
    const float* b_img = (const float*)d_in[16];
    const float* W_txt = (const float*)d_in[17];
    const float* b_txt = (const float*)d_in[18];

    long long nnz_adj = in_sizes[0];
    long long nnz_img = in_sizes[3];
    long long nnz_txt = in_sizes[6];

    float* out = (float*)d_out;
    const size_t N64 = (size_t)NTOT * D;
    const size_t I64 = (size_t)ITEM * D;

    float* Fimg = (float*)d_ws;      // ITEM x 64  (normed image feats)
    float* Ftxt = Fimg + I64;        // ITEM x 64  (normed text feats)
    float* T1   = Ftxt + I64;        // N x 64
    float* T2   = T1 + N64;          // N x 64
    float* T3   = T2 + N64;          // N x 64
    float* TB   = T3 + N64;          // N x 64
    float* TM   = TB + N64;          // N x 64

    const int EW_BLK = 256;
    const int ew_blocks = (int)((N64 + EW_BLK - 1) / EW_BLK);

    // ---- modality projections (WMMA fp32) + L2 normalize ----
    int gemm_blocks = ((ITEM / 16) + 7) / 8;   // 8 waves per 256-thread block
    gemm_wmma_f32<<<gemm_blocks, 256, 0, stream>>>(imgE, W_img, b_img, Fimg, ITEM, 1024);
    gemm_wmma_f32<<<gemm_blocks, 256, 0, stream>>>(txtE, W_txt, b_txt, Ftxt, ITEM, 768);
    l2norm_rows<<<(ITEM + 255) / 256, 256, 0, stream>>>(Fimg, ITEM);
    l2norm_rows<<<(ITEM + 255) / 256, 256, 0, stream>>>(Ftxt, ITEM);

    auto spmm = [&](const int* r, const int* c, const float* v, long long nnz,
                    const float* lo, const float* hi, float* y) {
        zero_kernel<<<2048, 256, 0, stream>>>(y, N64);
        long long threads = nnz * 16;
        int blocks = (int)((threads + 255) / 256);
        spmm_atomic<<<blocks, 256, 0, stream>>>(r, c, v, lo, hi, y, nnz);
    };

    // ---- image branch ----
    spmm(img_rows, img_cols, img_vals, nnz_img, uE, iE, T2);     // embedsImageAdj
    spmm(adj_rows, adj_cols, adj_vals, nnz_adj, uE, Fimg, T1);   // embedsImage
    spmm(adj_rows, adj_cols, adj_vals, nnz_adj, T1, iE, T3);     // embedsImage_
    fuse3_kernel<<<ew_blocks, EW_BLK, 0, stream>>>(T1, T3, T2, N64);  // T1 = final image embeds

    // ---- text branch ----
    spmm(txt_rows, txt_cols, txt_vals, nnz_txt, uE, iE, T2);     // embedsTextAdj
    spmm(adj_rows, adj_cols, adj_vals, nnz_adj, uE, Ftxt, TB);   // embedsText
    spmm(adj_rows, adj_cols, adj_vals, nnz_adj, TB, iE, T3);     // embedsText_
    fuse3_kernel<<<ew_blocks, EW_BLK, 0, stream>>>(TB, T3, T2, N64);  // TB = final text embeds

    // ---- modal fusion ----
    modal_fuse<<<ew_blocks, EW_BLK, 0, stream>>>(att_image, att_text, T1, TB, TM, N64);

    // ---- GCN layers with layer-sum aggregation (acc lives in d_out) ----
    hipMemcpyAsync(out, TM, N64 * sizeof(float), hipMemcpyDeviceToDevice, stream);
    spmm(adj_rows, adj_cols, adj_vals, nnz_adj, TM, TM + (size_t)USER * D, T3);
    add_inplace<<<ew_blocks, EW_BLK, 0, stream>>>(out, T3, N64);
    spmm(adj_rows, adj_cols, adj_vals, nnz_adj, T3, T3 + (size_t)USER * D, T1);
    add_inplace<<<ew_blocks, EW_BLK, 0, stream>>>(out, T1, N64);

    // ---- out += 0.5 * l2norm(embedsModal) ----
    final_addnorm<<<(NTOT + 255) / 256, 256, 0, stream>>>(TM, out, NTOT);
}